// PartResidualProjector_62191126446315
// MI455X (gfx1250) — compile-verified
//
#include <hip/hip_runtime.h>
#include <hip/hip_bf16.h>

typedef float v2f __attribute__((ext_vector_type(2)));
typedef float v8f __attribute__((ext_vector_type(8)));

#define BATCH 8
#define NPTS 4096
#define PRIM 16
#define HDIM 128
#define HID 128
#define RDIM 64
#define FOURH 512
#define NCHUNK 32
#define CHUNK (NPTS / NCHUNK)   // 128 points per chunk
#define KT 32                   // LDS sub-tile (points)
#define NTILES (CHUNK / KT)     // 4 sub-tiles per chunk
#define PART_STRIDE 6144        // 3 * 16*128 floats per chunk
#define EPSV 1e-6f

// Async DMA: global -> LDS, 16 bytes per lane, tracked by ASYNCcnt.
__device__ __forceinline__ void async_copy_b128(unsigned lds_addr, const void* gptr) {
  asm volatile("global_load_async_to_lds_b128 %0, %1, off"
               :: "v"(lds_addr), "v"((unsigned long long)(uintptr_t)gptr)
               : "memory");
}
__device__ __forceinline__ void wait_async0() {
  asm volatile("s_wait_asynccnt 0x0" ::: "memory");
}

// ---------------------------------------------------------------------------
// Kernel 1: per-chunk partial reductions, double-buffered async LDS staging.
//   accP[p][h] += a[n][p] * x[n][h]      (WMMA f32 16x16x4)
//   accS[p][h] += a[n][p] * x[n][h]^2    (WMMA f32 16x16x4, shared A operand)
//   mx[p][h]    = max over n (a>eps) of a[n][p]*x[n][h]   (VALU, b128 LDS reads)
// ---------------------------------------------------------------------------
__global__ __launch_bounds__(256) void prp_partial_kernel(
    const float* __restrict__ point, const float* __restrict__ assign,
    float* __restrict__ ws)
{
  __shared__ __align__(16) float s_pt[2][KT][HDIM];  // 2 x 16 KB
  __shared__ __align__(16) float s_as[2][KT][PRIM];  // 2 x  2 KB

  const int b    = blockIdx.x / NCHUNK;
  const int c    = blockIdx.x % NCHUNK;
  const int n0   = c * CHUNK;
  const int t    = threadIdx.x;
  const int lane = t & 31;
  const int wave = t >> 5;          // 0..7 -> owns h-tile [16w, 16w+16)
  const int h0   = wave * 16;
  const int col  = lane & 15;       // M index for A / N index for B
  const int kb   = (lane >> 4) * 2; // K sub-offset per ISA A-layout

  v8f accP = {};
  v8f accS = {};

  // max-phase mapping: thread owns (p = t&15, h in [8*(t>>4), 8*(t>>4)+8))
  const int mp = t & 15;
  const int hb = (t >> 4) * 8;
  float mx[8];
#pragma unroll
  for (int j = 0; j < 8; ++j) mx[j] = -__builtin_inff();

  const float* ptB = point  + (size_t)b * NPTS * HDIM;
  const float* asB = assign + (size_t)b * NPTS * PRIM;

  const unsigned pt0 = (unsigned)(uintptr_t)&s_pt[0][0][0];
  const unsigned as0 = (unsigned)(uintptr_t)&s_as[0][0][0];

  auto issue_tile = [&](int tile, int buf) {
    const int base = n0 + tile * KT;
    const float* gpt = ptB + (size_t)base * HDIM;       // KT*HDIM = 4096 floats
    const unsigned ptl = pt0 + (unsigned)buf * (KT * HDIM * 4);
#pragma unroll
    for (int i = 0; i < 4; ++i) {
      const unsigned j = (unsigned)t + 256u * i;        // float4 index
      async_copy_b128(ptl + j * 16u, gpt + j * 4);
    }
    if (t < 128) {                                      // KT*PRIM = 512 floats
      const float* gas = asB + (size_t)base * PRIM;
      const unsigned asl = as0 + (unsigned)buf * (KT * PRIM * 4);
      async_copy_b128(asl + (unsigned)t * 16u, gas + t * 4);
    }
  };

  issue_tile(0, 0);

  for (int it = 0; it < NTILES; ++it) {
    const int cur = it & 1;
    wait_async0();        // my async ops for tile 'it' complete (in-order)
    __syncthreads();      // whole tile resident; prior compute finished everywhere

    if (it + 1 < NTILES) issue_tile(it + 1, cur ^ 1);   // overlap with compute

    // WMMA phase: K steps of 4 points
#pragma unroll
    for (int k0 = 0; k0 < KT; k0 += 4) {
      v2f a, bp, bs;
      a.x = s_as[cur][k0 + kb][col];
      a.y = s_as[cur][k0 + kb + 1][col];
      const float p0 = s_pt[cur][k0 + kb][h0 + col];
      const float p1 = s_pt[cur][k0 + kb + 1][h0 + col];
      bp.x = p0;      bp.y = p1;
      bs.x = p0 * p0; bs.y = p1 * p1;
      accP = __builtin_amdgcn_wmma_f32_16x16x4_f32(false, a, false, bp,
                                                   (short)0, accP, false, false);
      accS = __builtin_amdgcn_wmma_f32_16x16x4_f32(false, a, false, bs,
                                                   (short)0, accS, false, false);
    }

    // masked-max phase (vectorized LDS reads: 2 x b128 per point)
    for (int k = 0; k < KT; ++k) {
      const float av = s_as[cur][k][mp];
      if (av > EPSV) {
        const float4 x0 = *(const float4*)&s_pt[cur][k][hb];
        const float4 x1 = *(const float4*)&s_pt[cur][k][hb + 4];
        mx[0] = fmaxf(mx[0], av * x0.x);
        mx[1] = fmaxf(mx[1], av * x0.y);
        mx[2] = fmaxf(mx[2], av * x0.z);
        mx[3] = fmaxf(mx[3], av * x0.w);
        mx[4] = fmaxf(mx[4], av * x1.x);
        mx[5] = fmaxf(mx[5], av * x1.y);
        mx[6] = fmaxf(mx[6], av * x1.z);
        mx[7] = fmaxf(mx[7], av * x1.w);
      }
    }
  }

  // write partials: D-layout => VGPR r holds rows (r, r+8), N = lane&15
  float* outP = ws + (size_t)blockIdx.x * PART_STRIDE;
  float* outS = outP + PRIM * HDIM;
  float* outM = outP + 2 * PRIM * HDIM;
#pragma unroll
  for (int r = 0; r < 8; ++r) {
    const int mm = r + 8 * (lane >> 4);
    outP[mm * HDIM + h0 + col] = accP[r];
    outS[mm * HDIM + h0 + col] = accS[r];
  }
  *(float4*)&outM[mp * HDIM + hb]     = make_float4(mx[0], mx[1], mx[2], mx[3]);
  *(float4*)&outM[mp * HDIM + hb + 4] = make_float4(mx[4], mx[5], mx[6], mx[7]);
}

// ---------------------------------------------------------------------------
// Kernel 2: reduce chunk partials, compute mass, assemble X (128 x 512).
// One block per (b,p) row, 128 threads = h.
// ---------------------------------------------------------------------------
__global__ __launch_bounds__(128) void prp_finalize_kernel(
    const float* __restrict__ sq, const float* __restrict__ assign,
    const float* __restrict__ part, float* __restrict__ X)
{
  const int row = blockIdx.x;           // b*16 + p
  const int b = row >> 4, p = row & 15;
  const int h = threadIdx.x;

  __shared__ float sred[128];
  const float* asB = assign + (size_t)b * NPTS * PRIM;
  float msum = 0.f;
  for (int n = h; n < NPTS; n += 128) msum += asB[(size_t)n * PRIM + p];
  sred[h] = msum;
  __syncthreads();
  for (int s = 64; s > 0; s >>= 1) {
    if (h < s) sred[h] += sred[h + s];
    __syncthreads();
  }
  const float mass = fmaxf(sred[0], EPSV);

  float pool = 0.f, sqm = 0.f, mxv = -__builtin_inff();
  for (int g = 0; g < NCHUNK; ++g) {
    const float* pg = part + (size_t)(b * NCHUNK + g) * PART_STRIDE;
    pool += pg[p * HDIM + h];
    sqm  += pg[PRIM * HDIM + p * HDIM + h];
    mxv  = fmaxf(mxv, pg[2 * PRIM * HDIM + p * HDIM + h]);
  }
  const float pooled = pool / mass;
  const float var    = sqm / mass - pooled * pooled;
  const float mxc    = (mxv == -__builtin_inff()) ? 0.f : mxv;

  float* xr = X + (size_t)row * FOURH;
  xr[h]               = sq[(size_t)row * HDIM + h];
  xr[HDIM + h]        = pooled;
  xr[2 * HDIM + h]    = mxc;
  xr[3 * HDIM + h]    = var;
}

// ---------------------------------------------------------------------------
// Kernel 3: MLP. relu(X(128x512)@W1 + b1) @ W2 + b2, all via WMMA f32.
// One WG, 8 waves; wave w owns M-tile [16w,16w+16).
// ---------------------------------------------------------------------------
__global__ __launch_bounds__(256) void prp_mlp_kernel(
    const float* __restrict__ X,  const float* __restrict__ W1,
    const float* __restrict__ b1, const float* __restrict__ W2,
    const float* __restrict__ b2, float* __restrict__ out)
{
  __shared__ float hid[HID * HID];  // 64 KB
  const int t    = threadIdx.x;
  const int lane = t & 31;
  const int wave = t >> 5;
  const int m0   = wave * 16;
  const int col  = lane & 15;
  const int kb   = (lane >> 4) * 2;

  // GEMM1: (128 x 512) @ (512 x 128)
  for (int nt = 0; nt < 8; ++nt) {
    const int n0 = nt * 16;
    v8f acc = {};
    for (int k0 = 0; k0 < FOURH; k0 += 4) {
      v2f a, bb;
      a.x  = X[(size_t)(m0 + col) * FOURH + k0 + kb];
      a.y  = X[(size_t)(m0 + col) * FOURH + k0 + kb + 1];
      bb.x = W1[(size_t)(k0 + kb) * HID + n0 + col];
      bb.y = W1[(size_t)(k0 + kb + 1) * HID + n0 + col];
      acc = __builtin_amdgcn_wmma_f32_16x16x4_f32(false, a, false, bb,
                                                  (short)0, acc, false, false);
    }
    const float bias = b1[n0 + col];
#pragma unroll
    for (int r = 0; r < 8; ++r) {
      const int mm = m0 + r + 8 * (lane >> 4);
      hid[mm * HID + n0 + col] = fmaxf(acc[r] + bias, 0.f);
    }
  }
  __syncthreads();

  // GEMM2: (128 x 128) @ (128 x 64)
  for (int nt = 0; nt < 4; ++nt) {
    const int n0 = nt * 16;
    v8f acc = {};
    for (int k0 = 0; k0 < HID; k0 += 4) {
      v2f a, bb;
      a.x  = hid[(m0 + col) * HID + k0 + kb];
      a.y  = hid[(m0 + col) * HID + k0 + kb + 1];
      bb.x = W2[(size_t)(k0 + kb) * RDIM + n0 + col];
      bb.y = W2[(size_t)(k0 + kb + 1) * RDIM + n0 + col];
      acc = __builtin_amdgcn_wmma_f32_16x16x4_f32(false, a, false, bb,
                                                  (short)0, acc, false, false);
    }
    const float bias = b2[n0 + col];
#pragma unroll
    for (int r = 0; r < 8; ++r) {
      const int mm = m0 + r + 8 * (lane >> 4);
      out[(size_t)mm * RDIM + n0 + col] = acc[r] + bias;
    }
  }
}

extern "C" void kernel_launch(void* const* d_in, const int* in_sizes, int n_in,
                              void* d_out, int out_size, void* d_ws, size_t ws_size,
                              hipStream_t stream) {
  const float* sq = (const float*)d_in[0];
  const float* pt = (const float*)d_in[1];
  const float* as = (const float*)d_in[2];
  const float* W1 = (const float*)d_in[3];
  const float* b1 = (const float*)d_in[4];
  const float* W2 = (const float*)d_in[5];
  const float* b2 = (const float*)d_in[6];
  float* out = (float*)d_out;
  float* ws  = (float*)d_ws;
  float* X   = ws + (size_t)BATCH * NCHUNK * PART_STRIDE;  // 128 x 512 staging

  prp_partial_kernel<<<BATCH * NCHUNK, 256, 0, stream>>>(pt, as, ws);
  prp_finalize_kernel<<<BATCH * PRIM, 128, 0, stream>>>(sq, as, ws, X);
  prp_mlp_kernel<<<1, 256, 0, stream>>>(X, W1, b1, W2, b2, out);
}